// RecoveryPolicy_2284922601915
// MI455X (gfx1250) — compile-verified
//
#include <hip/hip_runtime.h>
#include <math.h>

typedef __attribute__((ext_vector_type(2))) float v2f;
typedef __attribute__((ext_vector_type(8))) float v8f;

#define LOG2E   1.4426950408889634f
#define LOG2PI  1.8378770664093453f   // ln(2*pi)
#define B_TOT   4096
#define KK      128
#define CC      64
#define TILES_PER_WAVE 4

// ---------------------------------------------------------------------------
// Precompute per-(k,c) GEMM operands, written directly in WMMA lane layout.
//   Q[6,64] per k : coeffs of exponent poly in basis [1, z0, z1, z0^2, z0z1, z1^2]
//                   (rows 6,7 zero-padded; exp2-scaled, log-weight folded in)
//   W2[64,16] per k: cols [1, i00, i01, i00*m0+i01*m1, i10, i11, i10*m0+i11*m1, 0...]
//                   stored transposed as stage-2 WMMA A operand.
// ---------------------------------------------------------------------------
__global__ __launch_bounds__(256) void gmm_precompute_kernel(
    const float* __restrict__ weights, const float* __restrict__ means,
    const float* __restrict__ covs,
    float* __restrict__ qop,   // [K][4 j][2 h][32 lane][2]  = 512 floats / k
    float* __restrict__ w2op)  // [K][16 g][32 lane][2]      = 1024 floats / k
{
    int idx = blockIdx.x * blockDim.x + threadIdx.x;   // k*64 + c
    if (idx >= KK * CC) return;
    int k = idx >> 6, c = idx & 63;

    const float* cv = covs + (size_t)idx * 4;
    float a = cv[0], b = cv[1], cc2 = cv[2], d = cv[3];
    float det = a * d - b * cc2;
    float rdet = 1.0f / det;
    float i00 =  d * rdet, i01 = -b * rdet, i10 = -cc2 * rdet, i11 = a * rdet;
    float m0 = means[(size_t)idx * 2], m1 = means[(size_t)idx * 2 + 1];
    float w = weights[idx];

    // maha = A z0^2 + S z0 z1 + C z1^2 + D z0 + E z1 + F
    float S   = i01 + i10;
    float Dco = -(2.0f * i00 * m0 + S * m1);
    float Eco = -(S * m0 + 2.0f * i11 * m1);
    float Fco = i00 * m0 * m0 + S * m0 * m1 + i11 * m1 * m1;

    // exponent (base-2): L2E*(ln w - 0.5 ln det - ln 2pi - 0.5*maha)
    float q[8];
    q[0] = LOG2E * (logf(w) - 0.5f * logf(det) - LOG2PI - 0.5f * Fco);
    q[1] = -0.5f * LOG2E * Dco;   // * z0
    q[2] = -0.5f * LOG2E * Eco;   // * z1
    q[3] = -0.5f * LOG2E * i00;   // * z0^2
    q[4] = -0.5f * LOG2E * S;     // * z0 z1
    q[5] = -0.5f * LOG2E * i11;   // * z1^2
    q[6] = 0.0f; q[7] = 0.0f;

    // Scatter Q into stage-1 B-operand layout:
    // element (Krow=qi, N=c): j=c>>4, h=qi>>2, r=qi&3, lane=(c&15)+16*(r>>1), xy=r&1
    {
        int j = c >> 4;
        float* qk = qop + (((size_t)k * 4 + j) * 2) * 64;  // 2 h * 32 lane * 2
        #pragma unroll
        for (int qi = 0; qi < 8; ++qi) {
            int h = qi >> 2, r = qi & 3;
            int lane = (c & 15) + 16 * (r >> 1);
            qk[(h * 32 + lane) * 2 + (r & 1)] = q[qi];
        }
    }

    // W2 columns for the stage-2 reduction GEMM
    float w2[16];
    w2[0] = 1.0f;
    w2[1] = i00; w2[2] = i01; w2[3] = i00 * m0 + i01 * m1;
    w2[4] = i10; w2[5] = i11; w2[6] = i10 * m0 + i11 * m1;
    #pragma unroll
    for (int oc = 7; oc < 16; ++oc) w2[oc] = 0.0f;

    // Scatter W2^T into stage-2 A-operand layout:
    // element (row=oc, Kcol=c): g=c>>2, r=c&3, lane=oc+16*(r>>1), xy=r&1
    {
        int g = c >> 2, r = c & 3, xy = r & 1, hh = r >> 1;
        float* wk = w2op + (size_t)k * 1024;
        #pragma unroll
        for (int oc = 0; oc < 16; ++oc) {
            wk[(g * 32 + (oc + 16 * hh)) * 2 + xy] = w2[oc];
        }
    }
}

// ---------------------------------------------------------------------------
// Main kernel: block = one k, 8 wave32; each wave loops over 4 tiles of 16
// batch rows (per-k WMMA operands loaded once, amortized over 96 WMMAs).
// Per tile:
//   Stage 1:  E[16,64] = Zbasis[16,8] @ Q[8,64]      (8x v_wmma_f32_16x16x4_f32)
//   comp = exp2(E)                                    (32x v_exp_f32 / lane)
//   LDS transpose of comp (wave-private, stride 66 -> conflict-free;
//                          s_wait_dscnt 0 instead of block barrier)
//   Stage 2:  R[16,16] = W2^T[16,64] @ comp^T[64,16] (16x v_wmma_f32_16x16x4_f32)
//     -> lanes 0..15 each hold all 7 reductions for one batch row in registers.
// ---------------------------------------------------------------------------
__global__ __launch_bounds__(256) void gmm_wmma_kernel(
    const float* __restrict__ z, const float* __restrict__ scale,
    const float* __restrict__ qop, const float* __restrict__ w2op,
    float* __restrict__ out)
{
    __shared__ float lds[8][16][66];   // [wave][batch row][component], padded

    const int k    = blockIdx.x;
    const int wave = threadIdx.x >> 5;
    const int lane = threadIdx.x & 31;
    const int half = lane >> 4;
    const int m    = lane & 15;
    const int tileBase = (blockIdx.y * 8 + wave) * TILES_PER_WAVE;

    // ---- Per-k constant operands: load once, reuse for all tiles ----
    const v2f* qk = (const v2f*)(qop + (size_t)k * 512);
    v2f bq[4][2];
    #pragma unroll
    for (int j = 0; j < 4; ++j) {
        bq[j][0] = qk[(j * 2 + 0) * 32 + lane];
        bq[j][1] = qk[(j * 2 + 1) * 32 + lane];
    }
    const v2f* wk = (const v2f*)(w2op + (size_t)k * 1024);
    v2f aw[16];
    #pragma unroll
    for (int g = 0; g < 16; ++g) aw[g] = wk[g * 32 + lane];

    const float2 sc = ((const float2*)scale)[k];

    #pragma unroll 1
    for (int t = 0; t < TILES_PER_WAVE; ++t) {
        const int row = (tileBase + t) * 16 + m;

        const float2 zv = ((const float2*)z)[(size_t)row * KK + k];  // z:[B,K,2]
        const float z0 = zv.x, z1 = zv.y;

        // Stage-1 A operand: basis [1, z0, z1, z0^2, z0z1, z1^2, 0, 0]
        v2f aLo, aHi;
        aLo.x = half ? z1        : 1.0f;      // K2 | K0
        aLo.y = half ? z0 * z0   : z0;        // K3 | K1
        aHi.x = half ? 0.0f      : z0 * z1;   // K6 | K4
        aHi.y = half ? 0.0f      : z1 * z1;   // K7 | K5

        // ---- Stage 1 + exp2 + LDS staging ----
        #pragma unroll
        for (int j = 0; j < 4; ++j) {
            v8f e = {0.f,0.f,0.f,0.f,0.f,0.f,0.f,0.f};
            e = __builtin_amdgcn_wmma_f32_16x16x4_f32(false, aHi, false, bq[j][1],
                                                      (short)0, e, false, false);
            e = __builtin_amdgcn_wmma_f32_16x16x4_f32(false, aLo, false, bq[j][0],
                                                      (short)0, e, false, false);
            #pragma unroll
            for (int i = 0; i < 8; ++i) {
                // D layout: vgpr i, this lane -> E[Mrow = i + 8*half][c = 16*j + m]
                lds[wave][i + 8 * half][16 * j + m] = __builtin_amdgcn_exp2f(e[i]);
            }
        }
        // Wave-private LDS region: same-wave DS ops are in-order; only need the
        // split dep-counter wait (no block barrier).
        asm volatile("s_wait_dscnt 0" ::: "memory");

        // ---- Stage 2: reduction GEMM, B operand gathered from LDS ----
        v8f r = {0.f,0.f,0.f,0.f,0.f,0.f,0.f,0.f};
        #pragma unroll
        for (int g = 0; g < 16; ++g) {
            // B(Kc, N=m): lane -> {comp[m][4g + 2*half], comp[m][4g+1 + 2*half]}
            v2f bc = *reinterpret_cast<const v2f*>(&lds[wave][m][4 * g + 2 * half]);
            r = __builtin_amdgcn_wmma_f32_16x16x4_f32(false, aw[g], false, bc,
                                                      (short)0, r, false, false);
        }
        // WAR guard: next iteration's LDS writes must not pass these reads
        // (same-wave DS ordering is in-order, but keep the compiler honest).
        asm volatile("" ::: "memory");

        // ---- Epilogue: lanes 0..15 hold R[row][0..7] across the 8 acc VGPRs ----
        if (half == 0) {
            float pdf = r[0];
            float u0 = r[1], u1 = r[2], u2 = r[3];
            float v0 = r[4], v1 = r[5], v2 = r[6];
            float g0 = u2 - u0 * z0 - u1 * z1;    // = -sum comp * s0
            float g1 = v2 - v0 * z0 - v1 * z1;
            float gr0 = g0 * sc.x, gr1 = g1 * sc.y;
            float nrm = sqrtf(gr0 * gr0 + gr1 * gr1);
            float mag = __builtin_amdgcn_exp2f((5500.0f - nrm) * (LOG2E / 1100.0f));
            float s = mag / nrm;
            // sigmoid((pdf - 0.5)/0.5)
            float dens = 1.0f / (1.0f + __builtin_amdgcn_exp2f(-2.0f * LOG2E * (pdf - 0.5f)));

            float2* outD = (float2*)out;                              // [B,2K] as float2
            outD[(size_t)row * KK + k] = make_float2(dens, dens);
            float2* outG = (float2*)(out + (size_t)B_TOT * 2 * KK);
            outG[(size_t)row * KK + k] = make_float2(gr0 * s, gr1 * s);
        }
    }
}

extern "C" void kernel_launch(void* const* d_in, const int* in_sizes, int n_in,
                              void* d_out, int out_size, void* d_ws, size_t ws_size,
                              hipStream_t stream) {
    (void)in_sizes; (void)n_in; (void)out_size; (void)ws_size;
    const float* z       = (const float*)d_in[0];
    const float* weights = (const float*)d_in[1];
    const float* means   = (const float*)d_in[2];
    const float* covs    = (const float*)d_in[3];
    const float* scale   = (const float*)d_in[4];

    float* qop  = (float*)d_ws;            // 128 * 512  floats = 256 KB
    float* w2op = qop + (size_t)KK * 512;  // 128 * 1024 floats = 512 KB

    gmm_precompute_kernel<<<dim3(32), dim3(256), 0, stream>>>(weights, means, covs,
                                                              qop, w2op);
    gmm_wmma_kernel<<<dim3(KK, 8), dim3(256), 0, stream>>>(z, scale, qop, w2op,
                                                           (float*)d_out);
}